// GeneratorArchD_55448027791364
// MI455X (gfx1250) — compile-verified
//
#include <hip/hip_runtime.h>
#include <hip/hip_bf16.h>

// ---------------------------------------------------------------------------
// 4-qubit variational circuit, B = 2^20 samples.
// state-after-RX factorizes: amp(b) = m_b * (-i)^popcount(b), m_b real.
// Rest of circuit (CNOTs + RY) is a fixed real 16x16 matrix U(weights).
// => out = S^T |U_e r + i U_o im|^2 : two real 16x8 matvecs per sample.
// 16 samples batched into the N dimension of V_WMMA_F32_16X16X4_F32.
// All hot-loop state is explicit scalars (no local arrays) so nothing can be
// demoted to LDS/scratch by PromoteAlloca.
// ---------------------------------------------------------------------------

typedef float v2f __attribute__((ext_vector_type(2)));
typedef float v8f __attribute__((ext_vector_type(8)));

#define WAVES_PER_BLOCK 8
#define TILES_PER_WAVE  4   // 16 samples per tile -> 64 samples per wave

__device__ __forceinline__ v8f wmma_f32_16x16x4(v2f a, v2f b, v8f c) {
  // 8-arg form: (neg_a, A, neg_b, B, c_mod, C, reuse_a, reuse_b)
  return __builtin_amdgcn_wmma_f32_16x16x4_f32(false, a, false, b,
                                               (short)0, c, false, false);
}

// add value from lane^16 (SWAPX16 ds_swizzle: xor=0x10, and=0x1f -> 0x401F)
__device__ __forceinline__ float xadd16(float v) {
  int o = __builtin_amdgcn_ds_swizzle(__float_as_int(v), 0x401F);
  return v + __int_as_float(o);
}

// ---------------------------------------------------------------------------
// Pre-kernel: build the 16x16 real circuit matrix, column-permuted into
//   Up[r][0..7]  = U[r][even-popcount basis, ascending]   (acts on Re part)
//   Up[r][8..15] = U[r][odd-popcount basis, ascending]    (acts on Im part)
// One thread per basis column; evolve e_col through CNOTs + RY layers.
// Wire w <-> flat bit (3-w).
// ---------------------------------------------------------------------------
__global__ void build_unitary(const float* __restrict__ weights,
                              float* __restrict__ Up) {
  const int col = threadIdx.x;
  if (col >= 16) return;
  float v[16];
#pragma unroll
  for (int i = 0; i < 16; ++i) v[i] = (i == col) ? 1.0f : 0.0f;

#pragma unroll
  for (int l = 0; l < 2; ++l) {
    // all-to-all CNOTs, (c,t) with c < t, in reference order
#pragma unroll
    for (int c = 0; c < 4; ++c) {
#pragma unroll
      for (int t = 0; t < 4; ++t) {
        if (t > c) {
          const int cb = 1 << (3 - c), tb = 1 << (3 - t);
#pragma unroll
          for (int i = 0; i < 16; ++i) {
            if ((i & cb) && !(i & tb)) {
              const int j = i | tb;
              float tmp = v[i]; v[i] = v[j]; v[j] = tmp;
            }
          }
        }
      }
    }
    // RY(weights[l][w]) on each wire
#pragma unroll
    for (int w = 0; w < 4; ++w) {
      const float a  = weights[l * 4 + w] * 0.5f;
      const float ca = __cosf(a), sa = __sinf(a);
      const int wb = 1 << (3 - w);
#pragma unroll
      for (int i = 0; i < 16; ++i) {
        if (!(i & wb)) {
          const int j = i | wb;
          const float v0 = v[i], v1 = v[j];
          v[i] = ca * v0 - sa * v1;
          v[j] = sa * v0 + ca * v1;
        }
      }
    }
  }
  // packed column = parity*8 + rank of col among same-parity basis indices
  const int par = __popc(col) & 1;
  int rank = 0;
#pragma unroll
  for (int i = 0; i < 16; ++i)
    rank += ((i < col) && ((__popc(i) & 1) == par)) ? 1 : 0;
  const int pk = par * 8 + rank;
#pragma unroll
  for (int r = 0; r < 16; ++r) Up[r * 16 + pk] = v[r];
}

// ---------------------------------------------------------------------------
// Main kernel: 16 samples per WMMA tile.
//  A (16x4 f32): lane m (0..15) holds U row m; VGPR0 = K {0|2}, VGPR1 = K {1|3}
//                (low|high lane half) -> per chunk q: float2 at Up[m][4q+2h].
//  B (4x16 f32): lane n in half h supplies packed-basis rows {4q+2h, 4q+2h+1}
//                of the product state for sample n.
//  D (16x16 f32): VGPR v, lane (n,h) = amplitude of basis v+8h for sample n.
// ---------------------------------------------------------------------------
__global__ __launch_bounds__(32 * WAVES_PER_BLOCK)
void qsim_kernel(const float* __restrict__ x, const float* __restrict__ Up,
                 float* __restrict__ out, int batch) {
  const int lane = threadIdx.x & 31;
  const int wave = threadIdx.x >> 5;
  const int h    = lane >> 4;   // wave half
  const int n    = lane & 15;   // sample-in-tile == U row for A operand
  const bool hi  = (h != 0);

  const long long waveId     = (long long)blockIdx.x * WAVES_PER_BLOCK + wave;
  const long long sampleBase = waveId * (16LL * TILES_PER_WAVE);
  if (sampleBase >= batch) return;  // wave-uniform: EXEC stays all-ones

  // Load A operands once (L2-resident 1KB matrix): 2 K-chunks each for Re/Im.
  const float* urow = Up + n * 16;
  const v2f Ae0 = *(const v2f*)(urow + 0 + 2 * h);
  const v2f Ae1 = *(const v2f*)(urow + 4 + 2 * h);
  const v2f Ao0 = *(const v2f*)(urow + 8 + 2 * h);
  const v2f Ao1 = *(const v2f*)(urow + 12 + 2 * h);

#pragma unroll
  for (int t = 0; t < TILES_PER_WAVE; ++t) {
    const long long s = sampleBase + t * 16 + n;
    const float4 xv = *(const float4*)(x + s * 4);
    if (t + 1 < TILES_PER_WAVE)
      __builtin_prefetch(x + (s + 16) * 4, 0, 0);  // global_prefetch next tile

    // half-angle sincos per wire (wire w <-> flat bit 3-w) -- all scalars
    float s0, c0, s1, c1, s2, c2, s3, c3;
    __sincosf(xv.x * 0.5f, &s0, &c0);
    __sincosf(xv.y * 0.5f, &s1, &c1);
    __sincosf(xv.z * 0.5f, &s2, &c2);
    __sincosf(xv.w * 0.5f, &s3, &c3);

    // pair products: hi pair = wires 0,1 (bits 3,2); lo pair = wires 2,3
    const float cc = c0 * c1, cs = c0 * s1, sc = s0 * c1, ss = s0 * s1;
    const float CC = c2 * c3, CS = c2 * s3, SC = s2 * c3, SS = s2 * s3;

    // signed product-state values, packed even-popcount basis (Re support):
    //  b: 0(+) 3(-) 5(-) 6(-) 9(-) 10(-) 12(-) 15(+)   [(-i)^pc phase]
    const float r0 =   cc * CC;
    const float r1 = -(cc * SS);
    const float r2 = -(cs * CS);
    const float r3 = -(cs * SC);
    const float r4 = -(sc * CS);
    const float r5 = -(sc * SC);
    const float r6 = -(ss * CC);
    const float r7 =   ss * SS;
    // packed odd-popcount basis (Im support):
    //  b: 1(-) 2(-) 4(-) 7(+) 8(-) 11(+) 13(+) 14(+)
    const float i0 = -(cc * CS);
    const float i1 = -(cc * SC);
    const float i2 = -(cs * CC);
    const float i3 =   cs * SS;
    const float i4 = -(sc * CC);
    const float i5 =   sc * SS;
    const float i6 =   ss * CS;
    const float i7 =   ss * SC;

    // B operands: half h supplies packed rows {4q+2h, 4q+2h+1}
    v2f Bre0, Bre1, Bim0, Bim1;
    Bre0.x = hi ? r2 : r0;  Bre0.y = hi ? r3 : r1;
    Bre1.x = hi ? r6 : r4;  Bre1.y = hi ? r7 : r5;
    Bim0.x = hi ? i2 : i0;  Bim0.y = hi ? i3 : i1;
    Bim1.x = hi ? i6 : i4;  Bim1.y = hi ? i7 : i5;

    // two dense 16x8 matmuls (Re, Im) as 2+2 chained WMMAs over K-chunks
    v8f Dre = {0.f, 0.f, 0.f, 0.f, 0.f, 0.f, 0.f, 0.f};
    v8f Dim = {0.f, 0.f, 0.f, 0.f, 0.f, 0.f, 0.f, 0.f};
    Dre = wmma_f32_16x16x4(Ae0, Bre0, Dre);
    Dim = wmma_f32_16x16x4(Ao0, Bim0, Dim);
    Dre = wmma_f32_16x16x4(Ae1, Bre1, Dre);
    Dim = wmma_f32_16x16x4(Ao1, Bim1, Dim);

    // probabilities for this lane's 8 basis rows (b = v + 8h) -- scalars
    const float p0 = Dre[0] * Dre[0] + Dim[0] * Dim[0];
    const float p1 = Dre[1] * Dre[1] + Dim[1] * Dim[1];
    const float p2 = Dre[2] * Dre[2] + Dim[2] * Dim[2];
    const float p3 = Dre[3] * Dre[3] + Dim[3] * Dim[3];
    const float p4 = Dre[4] * Dre[4] + Dim[4] * Dim[4];
    const float p5 = Dre[5] * Dre[5] + Dim[5] * Dim[5];
    const float p6 = Dre[6] * Dre[6] + Dim[6] * Dim[6];
    const float p7 = Dre[7] * Dre[7] + Dim[7] * Dim[7];

    // <Z_w>: sign by bit (3-w) of b.  w=0 -> bit3 = h; w=1..3 -> bits of v.
    const float q01 = p0 + p1, q23 = p2 + p3, q45 = p4 + p5, q67 = p6 + p7;
    const float sAll = (q01 + q23) + (q45 + q67);
    const float part0 = hi ? -sAll : sAll;
    const float part1 = (q01 + q23) - (q45 + q67);                    // v & 4
    const float part2 = (q01 + q45) - (q23 + q67);                    // v & 2
    const float part3 = ((p0 + p2) + (p4 + p6)) -
                        ((p1 + p3) + (p5 + p7));                      // v & 1

    const float o0 = xadd16(part0);
    const float o1 = xadd16(part1);
    const float o2 = xadd16(part2);
    const float o3 = xadd16(part3);

    if (!hi) {
      float4 ov = make_float4(o0, o1, o2, o3);
      *(float4*)(out + s * 4) = ov;
    }
  }
}

// ---------------------------------------------------------------------------
extern "C" void kernel_launch(void* const* d_in, const int* in_sizes, int n_in,
                              void* d_out, int out_size, void* d_ws, size_t ws_size,
                              hipStream_t stream) {
  const float* x       = (const float*)d_in[0];   // [B,4] f32
  const float* weights = (const float*)d_in[1];   // [2,4] f32
  float* out = (float*)d_out;                     // [B,4] f32
  float* Up  = (float*)d_ws;                      // 16x16 f32 packed matrix

  const int batch = in_sizes[0] / 4;

  build_unitary<<<1, 16, 0, stream>>>(weights, Up);

  const int samplesPerBlock = 16 * TILES_PER_WAVE * WAVES_PER_BLOCK;  // 512
  const int blocks = (batch + samplesPerBlock - 1) / samplesPerBlock; // 2048
  qsim_kernel<<<blocks, 32 * WAVES_PER_BLOCK, 0, stream>>>(x, Up, out, batch);
}